// LeakDetector_86260123173277
// MI455X (gfx1250) — compile-verified
//
#include <hip/hip_runtime.h>
#include <math.h>

typedef __attribute__((ext_vector_type(2))) float v2f;
typedef __attribute__((ext_vector_type(8))) float v8f;

#define NEG_SLOPE 0.2f
#define BN_EPS 1e-5f

// ---------------------------------------------------------------------------
// Dense GEMM  H[M,Nout] = X[M,K] @ W[K,Nout]   (fp32, V_WMMA_F32_16X16X4_F32)
// One wave computes one 16x16 tile. M must be a multiple of 16 (50000 = 3125*16),
// Nout a multiple of 16, K a multiple of 4 (all hold here).
// A 16x4 fragment layout:  lane<16: (M=lane, K=k..k+1) in a.x/a.y
//                          lane>=16:(M=lane-16, K=k+2..k+3)
// B 4x16 fragment layout:  lane<16: (K=k..k+1, N=lane); lane>=16: K=k+2..k+3
// C/D 16x16: lane<16 holds M=0..7 (vgpr r), lane>=16 holds M=8..15, N=lane&15.
// ---------------------------------------------------------------------------
__global__ void gemm_wmma_f32(const float* __restrict__ X,
                              const float* __restrict__ W,
                              float* __restrict__ H,
                              int M, int K, int Nout) {
  const int lane  = threadIdx.x & 31;
  const int wave  = blockIdx.x * (blockDim.x >> 5) + (threadIdx.x >> 5);
  const int ntl   = Nout >> 4;
  const int mt    = wave / ntl;
  const int nt    = wave - mt * ntl;
  if (mt * 16 >= M) return;                 // wave-uniform early-out (EXEC stays all-1)
  const int l15   = lane & 15;
  const int khalf = (lane >> 4) << 1;       // 0 or 2
  const int row   = mt * 16 + l15;
  const int col   = nt * 16 + l15;
  const float* __restrict__ xr = X + (size_t)row * K;
  v8f c = {};
  for (int k = 0; k < K; k += 4) {
    v2f a, b;
    a.x = xr[k + khalf];
    a.y = xr[k + khalf + 1];
    b.x = W[(size_t)(k + khalf)     * Nout + col];
    b.y = W[(size_t)(k + khalf + 1) * Nout + col];
    c = __builtin_amdgcn_wmma_f32_16x16x4_f32(false, a, false, b,
                                              (short)0, c, false, false);
  }
  const int rbase = mt * 16 + ((lane >> 4) << 3);
#pragma unroll
  for (int r = 0; r < 8; ++r)
    H[(size_t)(rbase + r) * Nout + col] = c[r];
}

// ---------------------------------------------------------------------------
// Per-node attention scores: a_src[n,h] = <h[n,h,:], att_src[h,:]>, same for dst
// ---------------------------------------------------------------------------
__global__ void att_scores(const float* __restrict__ H,
                           const float* __restrict__ att_src,
                           const float* __restrict__ att_dst,
                           float* __restrict__ asrc, float* __restrict__ adst,
                           int N, int heads, int C) {
  int i = blockIdx.x * blockDim.x + threadIdx.x;
  if (i >= N * heads) return;
  int n = i / heads, h = i - n * heads;
  const float* hv = H + ((size_t)n * heads + h) * C;
  float s = 0.f, d = 0.f;
  for (int c = 0; c < C; ++c) {
    float v = hv[c];
    s += v * att_src[h * C + c];
    d += v * att_dst[h * C + c];
  }
  asrc[i] = s;
  adst[i] = d;
}

__device__ __forceinline__ void get_edge(const int* __restrict__ ei, int E,
                                         int e, int& s, int& d) {
  if (e < E) { s = ei[e]; d = ei[E + e]; }  // row 0 = src, row 1 = dst
  else       { s = d = e - E; }             // self loops appended
}

__device__ __forceinline__ void atomicMaxF(float* addr, float value) {
  if (value >= 0.f) atomicMax((int*)addr, __float_as_int(value));
  else              atomicMin((unsigned int*)addr, __float_as_uint(value));
}

// Pass 1: per-dst max of leaky_relu(a_src[src]+a_dst[dst])
__global__ void edge_max(const int* __restrict__ ei, int E, int Nn,
                         const float* __restrict__ asrc,
                         const float* __restrict__ adst,
                         float* __restrict__ emax, int heads) {
  int i = blockIdx.x * blockDim.x + threadIdx.x;
  int total = (E + Nn) * heads;
  if (i >= total) return;
  int e = i / heads, h = i - e * heads;
  int s, d; get_edge(ei, E, e, s, d);
  float v = asrc[s * heads + h] + adst[d * heads + h];
  v = v > 0.f ? v : NEG_SLOPE * v;
  atomicMaxF(&emax[d * heads + h], v);
}

// Pass 2: w = exp(e - max), accumulate per-dst denominator, stash w per edge
__global__ void edge_expsum(const int* __restrict__ ei, int E, int Nn,
                            const float* __restrict__ asrc,
                            const float* __restrict__ adst,
                            const float* __restrict__ emax,
                            float* __restrict__ denom,
                            float* __restrict__ wbuf, int heads) {
  int i = blockIdx.x * blockDim.x + threadIdx.x;
  int total = (E + Nn) * heads;
  if (i >= total) return;
  int e = i / heads, h = i - e * heads;
  int s, d; get_edge(ei, E, e, s, d);
  float v = asrc[s * heads + h] + adst[d * heads + h];
  v = v > 0.f ? v : NEG_SLOPE * v;
  float w = __expf(v - emax[d * heads + h]);
  wbuf[i] = w;
  atomicAdd(&denom[d * heads + h], w);
}

// Pass 3: agg[dst,h,:] += alpha * h[src,h,:]   (float4 gather + 4 atomics)
__global__ void edge_aggregate(const int* __restrict__ ei, int E, int Nn,
                               const float* __restrict__ H,
                               const float* __restrict__ wbuf,
                               const float* __restrict__ denom,
                               float* __restrict__ agg, int heads, int C) {
  int i = blockIdx.x * blockDim.x + threadIdx.x;
  int cg = C >> 2;
  int total = (E + Nn) * heads * cg;
  if (i >= total) return;
  int c4 = i % cg;
  int t  = i / cg;
  int h  = t % heads;
  int e  = t / heads;
  int s, d; get_edge(ei, E, e, s, d);
  float alpha = wbuf[(size_t)e * heads + h] /
                (denom[d * heads + h] + 1e-16f);
  const float4 v = ((const float4*)(H + ((size_t)s * heads + h) * C))[c4];
  float* o = agg + ((size_t)d * heads + h) * C + (c4 << 2);
  atomicAdd(o + 0, v.x * alpha);
  atomicAdd(o + 1, v.y * alpha);
  atomicAdd(o + 2, v.z * alpha);
  atomicAdd(o + 3, v.w * alpha);
}

// out = (maybe relu)(agg + bias)
__global__ void combine(const float* __restrict__ agg,
                        const float* __restrict__ bias,
                        float* __restrict__ out, int total, int F, int doRelu) {
  int i = blockIdx.x * blockDim.x + threadIdx.x;
  if (i >= total) return;
  float v = agg[i] + bias[i % F];
  if (doRelu) v = fmaxf(v, 0.f);
  out[i] = v;
}

// BatchNorm: column sums / sumsq into stats[0..F) and stats[F..2F)
__global__ void bn_reduce(const float* __restrict__ X, float* __restrict__ stats,
                          int N, int F) {
  int c = threadIdx.x;      // blockDim.x == F
  float s = 0.f, s2 = 0.f;
  for (int r = blockIdx.x; r < N; r += gridDim.x) {
    float v = X[(size_t)r * F + c];
    s += v; s2 += v * v;
  }
  atomicAdd(&stats[c], s);
  atomicAdd(&stats[F + c], s2);
}

__global__ void bn_norm(float* __restrict__ X, const float* __restrict__ stats,
                        const float* __restrict__ gamma,
                        const float* __restrict__ beta, int N, int F) {
  int i = blockIdx.x * blockDim.x + threadIdx.x;
  if (i >= N * F) return;
  int c = i % F;
  float invN = 1.f / (float)N;
  float mu   = stats[c] * invN;
  float var  = stats[F + c] * invN - mu * mu;
  X[i] = gamma[c] * (X[i] - mu) * rsqrtf(var + BN_EPS) + beta[c];
}

static inline unsigned cdiv(long long a, int b) { return (unsigned)((a + b - 1) / b); }

extern "C" void kernel_launch(void* const* d_in, const int* in_sizes, int n_in,
                              void* d_out, int out_size, void* d_ws, size_t ws_size,
                              hipStream_t stream) {
  (void)n_in; (void)out_size; (void)ws_size;
  const float* x0 = (const float*)d_in[0];
  const int*   ei = (const int*)d_in[1];
  const int N  = in_sizes[0] / 128;   // 50000
  const int E  = in_sizes[1] / 2;     // 1600000
  const int EA = E + N;               // edges + self loops

  // workspace layout (floats)
  float* ws = (float*)d_ws;
  size_t off = 0;
  float* bufA  = ws + off; off += (size_t)N * 128;
  float* bufB  = ws + off; off += (size_t)N * 128;
  float* hbuf  = ws + off; off += (size_t)N * 128;
  float* agg   = ws + off; off += (size_t)N * 128;
  float* asrc  = ws + off; off += (size_t)N * 4;
  float* adst  = ws + off; off += (size_t)N * 4;
  float* emax  = ws + off; off += (size_t)N * 4;
  float* denom = ws + off; off += (size_t)N * 4;
  float* wbuf  = ws + off; off += (size_t)EA * 4;
  float* stats = ws + off; off += 256;

  // {param base idx, Fin, heads, C, relu, bn gamma idx (-1 = none)}
  const int cfg[6][6] = {
    { 2, 128, 4,  32, 1, 18},
    { 6, 128, 4,  32, 1, 20},
    {10, 128, 4,  32, 1, 22},
    {14, 128, 1,  32, 1, -1},
    {24,  32, 1,  32, 1, -1},
    {28,  32, 1, 128, 0, -1},
  };

  const float* xin = x0;
  for (int l = 0; l < 6; ++l) {
    const int base  = cfg[l][0];
    const int Fin   = cfg[l][1];
    const int heads = cfg[l][2];
    const int C     = cfg[l][3];
    const int relu  = cfg[l][4];
    const int bnIdx = cfg[l][5];
    const int Fout  = heads * C;

    const float* Wp = (const float*)d_in[base + 0];
    const float* As = (const float*)d_in[base + 1];
    const float* Ad = (const float*)d_in[base + 2];
    const float* Bp = (const float*)d_in[base + 3];
    float* dest = (l == 5) ? (float*)d_out : ((l & 1) ? bufB : bufA);

    // 1) dense projection via WMMA
    long long waves = (long long)(N / 16) * (Fout / 16);
    gemm_wmma_f32<<<cdiv(waves, 8), 256, 0, stream>>>(xin, Wp, hbuf, N, Fin, Fout);

    // 2) per-node attention logits
    att_scores<<<cdiv((long long)N * heads, 256), 256, 0, stream>>>(
        hbuf, As, Ad, asrc, adst, N, heads, C);

    // 3) segment softmax over dst
    hipMemsetAsync(emax, 0xFF, (size_t)N * heads * sizeof(float), stream);
    hipMemsetAsync(denom, 0,   (size_t)N * heads * sizeof(float), stream);
    long long eh = (long long)EA * heads;
    edge_max<<<cdiv(eh, 256), 256, 0, stream>>>(ei, E, N, asrc, adst, emax, heads);
    edge_expsum<<<cdiv(eh, 256), 256, 0, stream>>>(ei, E, N, asrc, adst, emax,
                                                   denom, wbuf, heads);

    // 4) weighted scatter-sum
    hipMemsetAsync(agg, 0, (size_t)N * Fout * sizeof(float), stream);
    long long ehc = eh * (C >> 2);
    edge_aggregate<<<cdiv(ehc, 256), 256, 0, stream>>>(ei, E, N, hbuf, wbuf,
                                                       denom, agg, heads, C);

    // 5) bias (+ relu)
    combine<<<cdiv((long long)N * Fout, 256), 256, 0, stream>>>(
        agg, Bp, dest, N * Fout, Fout, relu);

    // 6) batch norm
    if (bnIdx >= 0) {
      const float* g = (const float*)d_in[bnIdx];
      const float* b = (const float*)d_in[bnIdx + 1];
      hipMemsetAsync(stats, 0, 2 * Fout * sizeof(float), stream);
      bn_reduce<<<256, Fout, 0, stream>>>(dest, stats, N, Fout);
      bn_norm<<<cdiv((long long)N * Fout, 256), 256, 0, stream>>>(
          dest, stats, g, b, N, Fout);
    }
    xin = dest;
  }
}